// TriCL_18107582120276
// MI455X (gfx1250) — compile-verified
//
#include <hip/hip_runtime.h>

#define D 128

typedef __attribute__((ext_vector_type(2))) float v2f;
typedef __attribute__((ext_vector_type(8))) float v8f;

// ---------------------------------------------------------------------------
// Y[M x 128] = X[M x 128] @ W[128 x 128] (+bias) (+prelu)
// grid.x = M/16 (M is a multiple of 16), block = 256 = 8 wave32s.
// Wave w computes the 16x16 tile at columns [16w, 16w+16).
// Both operands are staged in LDS with wide coalesced float4 loads:
//   - A tile: 16x128, stride 132 (bank-conflict-free, 16B-aligned copies)
//   - W     : 128x128, stride 132 (half-waves read rows k and k+2 -> 8 banks
//             apart; b0/b1 rows are 132 dwords apart -> ds_load_2addr fusable)
// Compute: 32 chained V_WMMA_F32_16X16X4_F32 steps over K=128.
// LDS: 16*132*4 + 128*132*4 = 76032 B -> 4 workgroups (32 waves) per WGP.
// ---------------------------------------------------------------------------
__global__ __launch_bounds__(256) void k_gemm128(
    const float* __restrict__ X, const float* __restrict__ W,
    const float* __restrict__ bias, const float* __restrict__ alpha,
    float* __restrict__ Y, int hasBias, int act)
{
  const int lane = threadIdx.x & 31;
  const int wave = threadIdx.x >> 5;
  const int row0 = blockIdx.x << 4;
  const int col0 = wave << 4;
  const int half = lane >> 4;     // 0: lanes 0-15, 1: lanes 16-31
  const int l16  = lane & 15;

  const int STR = 132;            // 128 + 4 pad (132 % 64 == 4)
  __shared__ float As[16 * 132];
  __shared__ float Ws[128 * 132];

  // Cooperative staging, float4-wide, fully coalesced.
  {
    const float4* Xv = reinterpret_cast<const float4*>(X + (size_t)row0 * D);
    const float4* Wv = reinterpret_cast<const float4*>(W);
    for (int t = threadIdx.x; t < 16 * (D / 4); t += 256) {   // 512 float4
      const int r = t >> 5, c4 = t & 31;
      *reinterpret_cast<float4*>(&As[r * STR + c4 * 4]) = Xv[t];
    }
    for (int t = threadIdx.x; t < 128 * (D / 4); t += 256) {  // 4096 float4
      const int r = t >> 5, c4 = t & 31;
      *reinterpret_cast<float4*>(&Ws[r * STR + c4 * 4]) = Wv[t];
    }
  }
  __syncthreads();

  v8f acc = {};
  const float* arow = &As[l16 * STR];
  const float* wcol = &Ws[col0 + l16];
#pragma unroll
  for (int k0 = 0; k0 < D; k0 += 4) {
    const int ka = k0 + 2 * half;   // A: lanes 0-15 -> K=k0,k0+1; 16-31 -> k0+2,k0+3
    v2f a, b;
    a[0] = arow[ka];
    a[1] = arow[ka + 1];
    b[0] = wcol[ka * STR];          // B VGPR0: rows K (lo half) / K+2 (hi half)
    b[1] = wcol[(ka + 1) * STR];    // B VGPR1: rows K+1 / K+3
    acc = __builtin_amdgcn_wmma_f32_16x16x4_f32(
        false, a, false, b, (short)0, acc, false, false);
  }

  const float al = alpha[0];
#pragma unroll
  for (int r = 0; r < 8; ++r) {
    const int orow = row0 + r + 8 * half;   // C/D layout: VGPR r = row r / r+8
    const int ocol = col0 + l16;
    float v = acc[r];
    if (hasBias) v += bias[ocol];
    if (act) v = (v >= 0.0f) ? v : al * v;
    Y[(size_t)orow * D + ocol] = v;
  }
}

// ---------------------------------------------------------------------------
// Degrees with self-loops: Dn starts at 1 (private hyperedge per node),
// De[e<E]=0, De[e>=E]=1; then histogram the real incidences; then invert.
// ---------------------------------------------------------------------------
__global__ void k_deg_init(float* __restrict__ Dn, float* __restrict__ De,
                           int N, int E, int NE)
{
  const int i = blockIdx.x * blockDim.x + threadIdx.x;
  if (i < N) Dn[i] = 1.0f;
  if (i < NE) De[i] = (i < E) ? 0.0f : 1.0f;
}

__global__ void k_deg_accum(const int* __restrict__ ni, const int* __restrict__ ei,
                            float* __restrict__ Dn, float* __restrict__ De, int nnz)
{
  const int i = blockIdx.x * blockDim.x + threadIdx.x;
  if (i < nnz) {
    atomicAdd(&Dn[ni[i]], 1.0f);
    atomicAdd(&De[ei[i]], 1.0f);
  }
}

__global__ void k_invert(float* __restrict__ b, int n)
{
  const int i = blockIdx.x * blockDim.x + threadIdx.x;
  if (i < n) { const float v = b[i]; b[i] = (v > 0.0f) ? (1.0f / v) : 0.0f; }
}

// ---------------------------------------------------------------------------
// Segment scatter-add: one wave per incidence, 4 floats per lane (float4
// gather from L2-resident src, 4 f32 atomics into L2-resident dst).
// dir=0: node->edge (dst=e, src=n, scale=De_inv[e])
// dir=1: edge->node (dst=n, src=e, scale=Dn_inv[n])
// Incidences [nnz, nnz+N) are the implicit self loops (n=i-nnz, e=E+i-nnz).
// ---------------------------------------------------------------------------
__global__ void k_scatter(const float* __restrict__ src, float* __restrict__ dst,
                          const int* __restrict__ node_idx, const int* __restrict__ edge_idx,
                          const float* __restrict__ scale, int nnz, int tot, int E, int dir)
{
  const int wid  = (blockIdx.x * blockDim.x + threadIdx.x) >> 5;
  const int lane = threadIdx.x & 31;
  if (wid >= tot) return;
  int n, e;
  if (wid < nnz) { n = node_idx[wid]; e = edge_idx[wid]; }
  else           { n = wid - nnz;     e = E + (wid - nnz); }
  const int s_row = dir ? e : n;
  const int d_row = dir ? n : e;
  const float sc = scale[d_row];
  const float4 v = reinterpret_cast<const float4*>(src + (size_t)s_row * D)[lane];
  float* dp = dst + (size_t)d_row * D + lane * 4;
  atomicAdd(dp + 0, v.x * sc);
  atomicAdd(dp + 1, v.y * sc);
  atomicAdd(dp + 2, v.z * sc);
  atomicAdd(dp + 3, v.w * sc);
}

__global__ void k_fill_bias(float* __restrict__ buf, const float* __restrict__ bias, int n)
{
  const int i = blockIdx.x * blockDim.x + threadIdx.x;
  if (i < n) buf[i] = bias[i & (D - 1)];
}

__global__ void k_prelu(float* __restrict__ buf, const float* __restrict__ alpha, int n)
{
  const int i = blockIdx.x * blockDim.x + threadIdx.x;
  if (i < n) { const float v = buf[i]; buf[i] = (v >= 0.0f) ? v : alpha[0] * v; }
}

__global__ void k_copy(float* __restrict__ dst, const float* __restrict__ src, int n)
{
  const int i = blockIdx.x * blockDim.x + threadIdx.x;
  if (i < n) dst[i] = src[i];
}

__global__ void k_concat(float* __restrict__ dst, const float* __restrict__ a,
                         const float* __restrict__ b, int n)  // n = N*D
{
  const int i = blockIdx.x * blockDim.x + threadIdx.x;
  if (i < n) {
    const int row = i >> 7, c = i & (D - 1);
    dst[(size_t)row * 2 * D + c]     = a[i];
    dst[(size_t)row * 2 * D + D + c] = b[i];
  }
}

// ---------------------------------------------------------------------------
extern "C" void kernel_launch(void* const* d_in, const int* in_sizes, int n_in,
                              void* d_out, int out_size, void* d_ws, size_t ws_size,
                              hipStream_t stream)
{
  const float* x        = (const float*)d_in[0];
  const float* xx       = (const float*)d_in[1];
  const int*   node_idx = (const int*)d_in[2];
  const int*   edge_idx = (const int*)d_in[3];
  const float* alpha    = (const float*)d_in[6];
  const float* e1_Wn2e  = (const float*)d_in[7];
  const float* e1_bn2e  = (const float*)d_in[8];
  const float* e1_We2n  = (const float*)d_in[9];
  const float* e1_be2n  = (const float*)d_in[10];
  const float* e2_Wn2e  = (const float*)d_in[11];
  const float* e2_bn2e  = (const float*)d_in[12];
  const float* e2_We2n  = (const float*)d_in[13];
  const float* e2_be2n  = (const float*)d_in[14];
  const float* d1_We    = (const float*)d_in[15];
  const float* d1_Wx    = (const float*)d_in[16];
  const float* d1_b     = (const float*)d_in[17];
  const float* d2_We    = (const float*)d_in[18];
  const float* d2_Wx    = (const float*)d_in[19];
  const float* d2_b     = (const float*)d_in[20];

  const int N   = in_sizes[0] / D;                       // 50000
  const int NNZ = in_sizes[2];                           // 800000
  const int E   = (out_size - 8 * N * D) / (2 * D);      // 20000 (derived; device scalar unreadable)
  const int NE  = E + N;                                 // edges + self loops
  const int TOT = NNZ + N;                               // incidences + self loops

  // workspace layout (fp32)
  float* ws = (float*)d_ws;
  float* Dn_inv = ws;                 ws += N;
  float* De_inv = ws;                 ws += NE;
  float* gemm_tmp = ws;               ws += (size_t)NE * D;
  float* e_full1  = ws;               ws += (size_t)NE * D;
  float* e_full2  = ws;               ws += (size_t)NE * D;
  float* x_cur    = ws;               ws += (size_t)N * D;
  float* decA     = ws;               ws += (size_t)N * D;

  // output layout (flat concat, return order)
  float* out   = (float*)d_out;
  float* o_nn1 = out;
  float* o_n1  = out + (size_t)N * 2 * D;
  float* o_e1  = o_n1 + (size_t)N * D;
  float* o_n2  = o_e1 + (size_t)E * D;
  float* o_e2  = o_n2 + (size_t)N * D;
  float* o_x11 = o_e2 + (size_t)E * D;
  float* o_x21 = o_x11 + (size_t)N * D;
  float* o_x12 = o_x21 + (size_t)N * D;
  float* o_x22 = o_x12 + (size_t)N * D;

  auto gemm = [&](const float* X, const float* W, const float* bias, float* Y,
                  int M, int act) {
    k_gemm128<<<M / 16, 256, 0, stream>>>(X, W, bias ? bias : alpha, alpha, Y,
                                          bias != nullptr, act);
  };
  auto fill = [&](float* buf, const float* bias, int rows) {
    const int n = rows * D;
    k_fill_bias<<<(n + 255) / 256, 256, 0, stream>>>(buf, bias, n);
  };
  auto prelu = [&](float* buf, int rows) {
    const int n = rows * D;
    k_prelu<<<(n + 255) / 256, 256, 0, stream>>>(buf, alpha, n);
  };
  auto scatter = [&](const float* src, float* dst, const float* scale, int dir) {
    k_scatter<<<(TOT + 7) / 8, 256, 0, stream>>>(src, dst, node_idx, edge_idx,
                                                 scale, NNZ, TOT, E, dir);
  };

  // ---- degrees (self-loop aware) ----
  k_deg_init<<<(NE + 255) / 256, 256, 0, stream>>>(Dn_inv, De_inv, N, E, NE);
  k_deg_accum<<<(NNZ + 255) / 256, 256, 0, stream>>>(node_idx, edge_idx, Dn_inv, De_inv, NNZ);
  k_invert<<<(N + 255) / 256, 256, 0, stream>>>(Dn_inv, N);
  k_invert<<<(NE + 255) / 256, 256, 0, stream>>>(De_inv, NE);

  // ---- encoder ----
  auto encoder = [&](const float* x0, const float* Wn2e, const float* bn2e,
                     const float* We2n, const float* be2n, float* n_out, float* e_full) {
    const float* xin = x0;
    for (int l = 0; l < 2; ++l) {
      gemm(xin, Wn2e + (size_t)l * D * D, nullptr, gemm_tmp, N, 0);
      fill(e_full, bn2e + (size_t)l * D, NE);
      scatter(gemm_tmp, e_full, De_inv, 0);
      prelu(e_full, NE);                               // e = prelu(seg + b)
      gemm(e_full, We2n + (size_t)l * D * D, nullptr, gemm_tmp, NE, 0);
      float* xout = (l == 0) ? x_cur : n_out;
      fill(xout, be2n + (size_t)l * D, N);
      scatter(gemm_tmp, xout, Dn_inv, 1);
      prelu(xout, N);                                  // x = prelu(n)
      xin = xout;
    }
  };

  // ---- decoder ----
  auto decoder = [&](const float* nin, const float* e_full, const float* We,
                     const float* Wx, const float* b, float* outp) {
    const float* xin = nin;
    for (int l = 0; l < 2; ++l) {
      gemm(e_full, We + (size_t)l * D * D, nullptr, gemm_tmp, NE, 0);
      float* xout = (l == 0) ? decA : outp;
      gemm(xin, Wx + (size_t)l * D * D, b + (size_t)l * D, xout, N, 0);  // x@Wx + b
      scatter(gemm_tmp, xout, Dn_inv, 1);                                 // + m
      prelu(xout, N);
      xin = xout;
    }
  };

  encoder(x,  e1_Wn2e, e1_bn2e, e1_We2n, e1_be2n, o_n1, e_full1);
  encoder(xx, e2_Wn2e, e2_bn2e, e2_We2n, e2_be2n, o_n2, e_full2);

  k_copy<<<(E * D + 255) / 256, 256, 0, stream>>>(o_e1, e_full1, E * D);
  k_copy<<<(E * D + 255) / 256, 256, 0, stream>>>(o_e2, e_full2, E * D);
  k_concat<<<(N * D + 255) / 256, 256, 0, stream>>>(o_nn1, o_n1, o_n2, N * D);

  decoder(o_n1, e_full1, d1_We, d1_Wx, d1_b, o_x11);
  decoder(o_n2, e_full2, d2_We, d2_Wx, d2_b, o_x21);
  decoder(o_n2, e_full2, d1_We, d1_Wx, d1_b, o_x12);
  decoder(o_n1, e_full1, d2_We, d2_Wx, d2_b, o_x22);
}